// MPMSimulator_74174085202612
// MI455X (gfx1250) — compile-verified
//
#include <hip/hip_runtime.h>
#include <stdint.h>

#define RESN 64
#define FLATN (RESN * RESN * RESN)
#define NPB 256  // particles per block (8 wave32)

typedef float v2f __attribute__((ext_vector_type(2)));
typedef float v8f __attribute__((ext_vector_type(8)));

__device__ __forceinline__ int clampi(int v, int lo, int hi) {
    return v < lo ? lo : (v > hi ? hi : v);
}

// ---------------------------------------------------------------------------
// Tensor Data Mover: 1-D copy of `nwords` dwords from global -> LDS.
// D# built per CDNA5 ISA ch.8 (group0: flags/lds_addr/global_addr/type=2,
// group1: data_size=4B, tensor_dim0 = tile_dim0 = nwords, stride = nwords).
// ---------------------------------------------------------------------------
__device__ __forceinline__ void tdm_copy_words(uint32_t lds_addr,
                                               const void* gptr,
                                               uint32_t nwords) {
    typedef unsigned int v4u __attribute__((ext_vector_type(4)));
    typedef int          v4i __attribute__((ext_vector_type(4)));
    typedef int          v8i __attribute__((ext_vector_type(8)));

    uint64_t ga = (uint64_t)(size_t)gptr;

    v4u g0;
    g0.x = 1u;                                            // count=1 (valid), user mode
    g0.y = lds_addr;                                      // LDS byte address
    g0.z = (uint32_t)ga;                                  // global_addr[31:0]
    g0.w = ((uint32_t)(ga >> 32) & 0x01FFFFFFu)           // global_addr[56:32]
         | (2u << 30);                                    // type = 2 ("image")

    v8i g1;
    g1[0] = (int)(2u << 16);                              // data_size = 4 bytes
    g1[1] = (int)((nwords & 0xFFFFu) << 16);              // tensor_dim0[15:0]
    g1[2] = (int)(((nwords >> 16) & 0xFFFFu)              // tensor_dim0[31:16]
         | (1u << 16));                                   // tensor_dim1 = 1
    g1[3] = (int)((nwords & 0xFFFFu) << 16);              // tile_dim0 = nwords
    g1[4] = 0;                                            // tile_dim1/2 unused
    g1[5] = (int)nwords;                                  // tensor_dim0_stride[31:0]
    g1[6] = 0;
    g1[7] = 0;

    v4i z4 = {0, 0, 0, 0};
#if defined(__clang_major__) && (__clang_major__ >= 23)
    v8i z8 = {0, 0, 0, 0, 0, 0, 0, 0};
    __builtin_amdgcn_tensor_load_to_lds(g0, g1, z4, z4, z8, 0);
#else
    __builtin_amdgcn_tensor_load_to_lds(g0, g1, z4, z4, 0);
#endif
}

__device__ __forceinline__ void atomic_add_f32(float* p, float v) {
    __hip_atomic_fetch_add(p, v, __ATOMIC_RELAXED, __HIP_MEMORY_SCOPE_AGENT);
}

// ---------------------------------------------------------------------------
// Kernel 0: zero the grid (mass, mom.xyz) — 4 MB, L2-resident.
// ---------------------------------------------------------------------------
__global__ void mpm_zero_grid(float4* __restrict__ grid) {
    int id = blockIdx.x * blockDim.x + threadIdx.x;
    if (id < FLATN) grid[id] = make_float4(0.f, 0.f, 0.f, 0.f);
}

// ---------------------------------------------------------------------------
// Kernel 1: P2G scatter. Particle slab (24 KB) staged to LDS via TDM while
// the VMEM pipe is reserved for the 27-node atomic scatter of
// (mass, momentum) into the L2-resident grid.
// ---------------------------------------------------------------------------
__global__ void __launch_bounds__(NPB) mpm_p2g(
    const float* __restrict__ pos, const float* __restrict__ vel,
    const float* __restrict__ Fm,  const float* __restrict__ Cm,
    float* __restrict__ grid, int n) {

    __shared__ float s_pos[NPB * 3];
    __shared__ float s_vel[NPB * 3];
    __shared__ float s_F[NPB * 9];
    __shared__ float s_C[NPB * 9];

    const int start = blockIdx.x * NPB;
    const int cnt   = min(NPB, n - start);
    if (cnt <= 0) return;

    if (threadIdx.x < 32) {  // one wave drives the DMA engine (TDM ignores EXEC)
        tdm_copy_words((uint32_t)(size_t)(void*)s_pos, pos + (size_t)start * 3, (uint32_t)cnt * 3u);
        tdm_copy_words((uint32_t)(size_t)(void*)s_vel, vel + (size_t)start * 3, (uint32_t)cnt * 3u);
        tdm_copy_words((uint32_t)(size_t)(void*)s_F,   Fm  + (size_t)start * 9, (uint32_t)cnt * 9u);
        tdm_copy_words((uint32_t)(size_t)(void*)s_C,   Cm  + (size_t)start * 9, (uint32_t)cnt * 9u);
        __builtin_amdgcn_s_wait_tensorcnt(0);
    }
    __syncthreads();

    const int t = threadIdx.x;
    if (t >= cnt) return;

    const float DXf   = 1.0f / 64.0f;
    const float INVDX = 64.0f;
    const float DT    = 1e-4f;
    const float VOLc  = 0.5f * DXf * DXf * DXf;
    const float MU    = 10000.0f / (2.0f * 1.2f);
    const float LAM   = 10000.0f * 0.2f / (1.2f * 0.6f);
    const float MASS  = VOLc;                              // RHO = 1
    const float SCOEF = -DT * VOLc * 4.0f * INVDX * INVDX; // -dt*vol*4/dx^2

    const float p0 = s_pos[t * 3 + 0], p1 = s_pos[t * 3 + 1], p2 = s_pos[t * 3 + 2];
    const float v0 = s_vel[t * 3 + 0], v1 = s_vel[t * 3 + 1], v2 = s_vel[t * 3 + 2];
    float F0 = s_F[t * 9 + 0], F1 = s_F[t * 9 + 1], F2 = s_F[t * 9 + 2];
    float F3 = s_F[t * 9 + 3], F4 = s_F[t * 9 + 4], F5 = s_F[t * 9 + 5];
    float F6 = s_F[t * 9 + 6], F7 = s_F[t * 9 + 7], F8 = s_F[t * 9 + 8];
    float C0 = s_C[t * 9 + 0], C1 = s_C[t * 9 + 1], C2 = s_C[t * 9 + 2];
    float C3 = s_C[t * 9 + 3], C4 = s_C[t * 9 + 4], C5 = s_C[t * 9 + 5];
    float C6 = s_C[t * 9 + 6], C7 = s_C[t * 9 + 7], C8 = s_C[t * 9 + 8];

    // Signed cofactors cof[i][j]; inv(F)^T = cof / det.
    const float c00 = F4 * F8 - F5 * F7;
    const float c01 = F5 * F6 - F3 * F8;
    const float c02 = F3 * F7 - F4 * F6;
    const float c10 = F2 * F7 - F1 * F8;
    const float c11 = F0 * F8 - F2 * F6;
    const float c12 = F1 * F6 - F0 * F7;
    const float c20 = F1 * F5 - F2 * F4;
    const float c21 = F2 * F3 - F0 * F5;
    const float c22 = F0 * F4 - F1 * F3;
    const float det = F0 * c00 + F1 * c01 + F2 * c02;

    const float J    = fmaxf(det, 1e-8f);
    const float rdet = 1.0f / det;
    // P = MU*F + (LAM*(J-1) - MU) * F^-T
    const float k2 = (LAM * (J - 1.0f) - MU) * rdet;
    const float P0 = MU * F0 + k2 * c00, P1 = MU * F1 + k2 * c01, P2 = MU * F2 + k2 * c02;
    const float P3 = MU * F3 + k2 * c10, P4 = MU * F4 + k2 * c11, P5 = MU * F5 + k2 * c12;
    const float P6 = MU * F6 + k2 * c20, P7 = MU * F7 + k2 * c21, P8 = MU * F8 + k2 * c22;

    // affine = SCOEF * (P F^T) + MASS*C
    const float a00 = SCOEF * (P0 * F0 + P1 * F1 + P2 * F2) + MASS * C0;
    const float a01 = SCOEF * (P0 * F3 + P1 * F4 + P2 * F5) + MASS * C1;
    const float a02 = SCOEF * (P0 * F6 + P1 * F7 + P2 * F8) + MASS * C2;
    const float a10 = SCOEF * (P3 * F0 + P4 * F1 + P5 * F2) + MASS * C3;
    const float a11 = SCOEF * (P3 * F3 + P4 * F4 + P5 * F5) + MASS * C4;
    const float a12 = SCOEF * (P3 * F6 + P4 * F7 + P5 * F8) + MASS * C5;
    const float a20 = SCOEF * (P6 * F0 + P7 * F1 + P8 * F2) + MASS * C6;
    const float a21 = SCOEF * (P6 * F3 + P7 * F4 + P8 * F5) + MASS * C7;
    const float a22 = SCOEF * (P6 * F6 + P7 * F7 + P8 * F8) + MASS * C8;

    const float xp0 = p0 * INVDX - 0.5f, xp1 = p1 * INVDX - 0.5f, xp2 = p2 * INVDX - 0.5f;
    const float bf0 = floorf(xp0), bf1 = floorf(xp1), bf2 = floorf(xp2);
    const int bx = (int)bf0, by = (int)bf1, bz = (int)bf2;
    const float fx0 = xp0 - bf0, fx1 = xp1 - bf1, fx2 = xp2 - bf2;

    const float w0[3] = {0.5f * (1.5f - fx0) * (1.5f - fx0), 0.75f - (fx0 - 1.f) * (fx0 - 1.f), 0.5f * (fx0 - 0.5f) * (fx0 - 0.5f)};
    const float w1[3] = {0.5f * (1.5f - fx1) * (1.5f - fx1), 0.75f - (fx1 - 1.f) * (fx1 - 1.f), 0.5f * (fx1 - 0.5f) * (fx1 - 0.5f)};
    const float w2[3] = {0.5f * (1.5f - fx2) * (1.5f - fx2), 0.75f - (fx2 - 1.f) * (fx2 - 1.f), 0.5f * (fx2 - 0.5f) * (fx2 - 0.5f)};

    const float mv0 = MASS * v0, mv1 = MASS * v1, mv2 = MASS * v2;

#pragma unroll
    for (int i = 0; i < 3; ++i) {
        const int nx = clampi(bx + i, 0, RESN - 1);
        const float dp0 = ((float)i - fx0) * DXf;
        const float wi = w0[i];
#pragma unroll
        for (int j = 0; j < 3; ++j) {
            const int ny = clampi(by + j, 0, RESN - 1);
            const float dp1 = ((float)j - fx1) * DXf;
            const float wij = wi * w1[j];
#pragma unroll
            for (int k = 0; k < 3; ++k) {
                const int nz = clampi(bz + k, 0, RESN - 1);
                const float dp2 = ((float)k - fx2) * DXf;
                const float ww = wij * w2[k];
                const int fi = (nx * RESN + ny) * RESN + nz;
                float* cell = grid + 4 * (size_t)fi;
                const float m0 = ww * (mv0 + a00 * dp0 + a01 * dp1 + a02 * dp2);
                const float m1 = ww * (mv1 + a10 * dp0 + a11 * dp1 + a12 * dp2);
                const float m2 = ww * (mv2 + a20 * dp0 + a21 * dp1 + a22 * dp2);
                atomic_add_f32(cell + 0, ww * MASS);
                atomic_add_f32(cell + 1, m0);
                atomic_add_f32(cell + 2, m1);
                atomic_add_f32(cell + 3, m2);
            }
        }
    }
}

// ---------------------------------------------------------------------------
// Kernel 2: grid normalize + gravity + boundary conditions (in place).
// ---------------------------------------------------------------------------
__global__ void mpm_grid_update(float4* __restrict__ grid) {
    int id = blockIdx.x * blockDim.x + threadIdx.x;
    if (id >= FLATN) return;
    float4 g = grid[id];
    float vx = 0.f, vy = 0.f, vz = 0.f;
    if (g.x > 1e-10f) {
        const float r = 1.0f / g.x;
        vx = g.y * r; vy = g.z * r; vz = g.w * r;
    }
    vy += -9.81f * 1e-4f;  // DT * gravity.y
    const int x = id >> 12, y = (id >> 6) & 63, z = id & 63;
    const bool bc = (x < 3) | (x >= RESN - 3) | (y < 3) | (y >= RESN - 3) |
                    (z < 3) | (z >= RESN - 3);
    if (bc) { vx = 0.f; vy = 0.f; vz = 0.f; }
    grid[id] = make_float4(vx, vy, vz, 0.f);
}

// ---------------------------------------------------------------------------
// Kernel 3: G2P gather + advect; F_new = (I + dt*C_new)·F evaluated on the
// matrix pipe via V_WMMA_F32_16X16X4_F32, block-diagonal batching of 5
// particles per tile (diagonal 3x3 blocks of D are the exact per-particle
// products; off-diagonal blocks are never read). No early lane returns:
// WMMA requires EXEC == all-1s, so tail lanes run with clamped indices and
// are masked only at the final store.
// ---------------------------------------------------------------------------
__global__ void __launch_bounds__(NPB) mpm_g2p(
    const float* __restrict__ pos, const float* __restrict__ Fm,
    const float4* __restrict__ gridv, float* __restrict__ out, int n) {

    __shared__ float s_pos[NPB * 3];
    __shared__ float s_F[NPB * 9];
    __shared__ float s_A[NPB * 9];               // A = I + dt*NC per particle
    __shared__ float s_D[(NPB / 32) * 256];      // one 16x16 D image per wave

    const int start = blockIdx.x * NPB;
    const int cnt   = min(NPB, n - start);
    if (cnt <= 0) return;

    if (threadIdx.x < 32) {
        tdm_copy_words((uint32_t)(size_t)(void*)s_pos, pos + (size_t)start * 3, (uint32_t)cnt * 3u);
        tdm_copy_words((uint32_t)(size_t)(void*)s_F,   Fm  + (size_t)start * 9, (uint32_t)cnt * 9u);
        __builtin_amdgcn_s_wait_tensorcnt(0);
    }
    __syncthreads();

    const int t    = threadIdx.x;
    const int lane = t & 31;
    const int wv   = t >> 5;
    const bool active = (t < cnt);

    const float DXf   = 1.0f / 64.0f;
    const float INVDX = 64.0f;
    const float DT    = 1e-4f;
    const float K4    = 4.0f * INVDX * INVDX;  // 16384

    const float p0 = s_pos[t * 3 + 0];
    const float p1 = s_pos[t * 3 + 1];
    const float p2 = s_pos[t * 3 + 2];

    const float xp0 = p0 * INVDX - 0.5f, xp1 = p1 * INVDX - 0.5f, xp2 = p2 * INVDX - 0.5f;
    const float bf0 = floorf(xp0), bf1 = floorf(xp1), bf2 = floorf(xp2);
    const int bx = (int)bf0, by = (int)bf1, bz = (int)bf2;
    const float fx0 = xp0 - bf0, fx1 = xp1 - bf1, fx2 = xp2 - bf2;

    const float w0[3] = {0.5f * (1.5f - fx0) * (1.5f - fx0), 0.75f - (fx0 - 1.f) * (fx0 - 1.f), 0.5f * (fx0 - 0.5f) * (fx0 - 0.5f)};
    const float w1[3] = {0.5f * (1.5f - fx1) * (1.5f - fx1), 0.75f - (fx1 - 1.f) * (fx1 - 1.f), 0.5f * (fx1 - 0.5f) * (fx1 - 0.5f)};
    const float w2[3] = {0.5f * (1.5f - fx2) * (1.5f - fx2), 0.75f - (fx2 - 1.f) * (fx2 - 1.f), 0.5f * (fx2 - 0.5f) * (fx2 - 0.5f)};

    float nv0 = 0.f, nv1 = 0.f, nv2 = 0.f;
    float NC[9] = {0.f, 0.f, 0.f, 0.f, 0.f, 0.f, 0.f, 0.f, 0.f};

#pragma unroll
    for (int i = 0; i < 3; ++i) {
        const int nx = clampi(bx + i, 0, RESN - 1);   // in-range even for tail lanes
        const float dp0 = ((float)i - fx0) * DXf;
        const float wi = w0[i];
#pragma unroll
        for (int j = 0; j < 3; ++j) {
            const int ny = clampi(by + j, 0, RESN - 1);
            const float dp1 = ((float)j - fx1) * DXf;
            const float wij = wi * w1[j];
#pragma unroll
            for (int k = 0; k < 3; ++k) {
                const int nz = clampi(bz + k, 0, RESN - 1);
                const float dp2 = ((float)k - fx2) * DXf;
                const float ww = wij * w2[k];
                const int fi = (nx * RESN + ny) * RESN + nz;
                const float4 g = gridv[fi];  // L2-resident gather
                nv0 += ww * g.x; nv1 += ww * g.y; nv2 += ww * g.z;
                const float s = K4 * ww;
                NC[0] += s * g.x * dp0; NC[1] += s * g.x * dp1; NC[2] += s * g.x * dp2;
                NC[3] += s * g.y * dp0; NC[4] += s * g.y * dp1; NC[5] += s * g.y * dp2;
                NC[6] += s * g.z * dp0; NC[7] += s * g.z * dp1; NC[8] += s * g.z * dp2;
            }
        }
    }

    // Stage A = I + DT*NC for the matrix pipe.
#pragma unroll
    for (int q = 0; q < 9; ++q) {
        const float eye = (q == 0 || q == 4 || q == 8) ? 1.0f : 0.0f;
        s_A[t * 9 + q] = eye + DT * NC[q];
    }
    __syncthreads();

    // ---- WMMA phase: F_new = A * F, 5 particles per 16x16x4 f32 tile ----
    const int half = lane >> 4;            // 0/1 -> K pair {0,1} / {2,3}
    const int lr   = lane & 15;            // row (A) / col (B) in tile
    const int qm   = lr / 3;               // particle slot within tile
    const int im   = lr - qm * 3;          // row/col within 3x3 block
    const float* sAw = s_A + wv * 32 * 9;
    const float* sFw = s_F + wv * 32 * 9;
    float*       sDw = s_D + wv * 256;

    float Fn[9] = {0.f, 0.f, 0.f, 0.f, 0.f, 0.f, 0.f, 0.f, 0.f};

    for (int grp = 0; grp < 7; ++grp) {    // ceil(32 / 5) groups per wave
        const int wp = grp * 5 + qm;       // wave-local particle for this row/col
        v2f a, b;
#pragma unroll
        for (int v = 0; v < 2; ++v) {
            const int K = half * 2 + v;
            const bool ok = (lr < 15) && (wp < 32) && (K < 3);
            a[v] = ok ? sAw[wp * 9 + im * 3 + K] : 0.0f;  // A[row=3q+im][K]
            b[v] = ok ? sFw[wp * 9 + K * 3 + im] : 0.0f;  // B[K][col=3q+im] = F[K][im]
        }
        v8f c = {0.f, 0.f, 0.f, 0.f, 0.f, 0.f, 0.f, 0.f};
        v8f d = __builtin_amdgcn_wmma_f32_16x16x4_f32(
            false, a, false, b, (short)0, c, false, false);

        __syncthreads();                   // protect previous group's D reads
#pragma unroll
        for (int v = 0; v < 8; ++v)        // D image: M = v + 8*half, N = lr
            sDw[(v + 8 * half) * 16 + lr] = d[v];
        __syncthreads();

        if (lane >= grp * 5 && lane < grp * 5 + 5) {  // my particle is in this tile
            const int base = 3 * (lane - grp * 5);
#pragma unroll
            for (int i = 0; i < 3; ++i)
#pragma unroll
                for (int j = 0; j < 3; ++j)
                    Fn[i * 3 + j] = sDw[(base + i) * 16 + (base + j)];
        }
    }

    if (!active) return;
    const int p = start + t;

    float o[24];
    o[0] = p0 + DT * nv0;
    o[1] = p1 + DT * nv1;
    o[2] = p2 + DT * nv2;
    o[3] = nv0; o[4] = nv1; o[5] = nv2;
#pragma unroll
    for (int q = 0; q < 9; ++q) o[6 + q]  = Fn[q];
#pragma unroll
    for (int q = 0; q < 9; ++q) o[15 + q] = NC[q];

    float4* op = (float4*)(out + (size_t)p * 24);  // 96 B/row, 16 B aligned
#pragma unroll
    for (int q = 0; q < 6; ++q)
        op[q] = make_float4(o[4 * q + 0], o[4 * q + 1], o[4 * q + 2], o[4 * q + 3]);
}

// ---------------------------------------------------------------------------
extern "C" void kernel_launch(void* const* d_in, const int* in_sizes, int n_in,
                              void* d_out, int out_size, void* d_ws, size_t ws_size,
                              hipStream_t stream) {
    const float* pos = (const float*)d_in[0];
    const float* vel = (const float*)d_in[1];
    const float* Fm  = (const float*)d_in[2];
    const float* Cm  = (const float*)d_in[3];
    // d_in[4] (Jp) unused by the reference step.
    const int n = in_sizes[4];

    float* grid = (float*)d_ws;  // FLAT * float4 = 4 MB scratch (L2-resident)
    float* out  = (float*)d_out;

    const int gblocks = (FLATN + 255) / 256;
    mpm_zero_grid<<<gblocks, 256, 0, stream>>>((float4*)grid);
    mpm_p2g<<<(n + NPB - 1) / NPB, NPB, 0, stream>>>(pos, vel, Fm, Cm, grid, n);
    mpm_grid_update<<<gblocks, 256, 0, stream>>>((float4*)grid);
    mpm_g2p<<<(n + NPB - 1) / NPB, NPB, 0, stream>>>(pos, Fm, (const float4*)grid, out, n);
}